// HybridModel_87806311399503
// MI455X (gfx1250) — compile-verified
//
#include <hip/hip_runtime.h>
#include <math.h>

#define NN 50000
#define NE 800000
#define LDS_STRIDE 66   // 64 cols + pad: avoids bank conflict between lane halves

typedef float v2f __attribute__((ext_vector_type(2)));
typedef float v8f __attribute__((ext_vector_type(8)));

__device__ __forceinline__ v8f wmma4(v2f a, v2f b, v8f c) {
  // V_WMMA_F32_16X16X4_F32 : D = A(16x4) * B(4x16) + C, exact fp32
  return __builtin_amdgcn_wmma_f32_16x16x4_f32(false, a, false, b, (short)0, c,
                                               false, false);
}

// async global->LDS copy of one dword per lane (ASYNCcnt-tracked)
__device__ __forceinline__ void async_copy_b32(const float* gptr, const float* lptr) {
  unsigned lds = (unsigned)(size_t)lptr;                   // flat LDS addr low 32b = LDS offset
  unsigned long long ga = (unsigned long long)(size_t)gptr;
  asm volatile("global_load_async_to_lds_b32 %0, %1, off" ::"v"(lds), "v"(ga)
               : "memory");
}
__device__ __forceinline__ void async_wait0() {
  asm volatile("s_wait_asynccnt 0x0" ::: "memory");
}

// ---- ordered-uint mapping for atomic float max ----
__device__ __forceinline__ unsigned f2ord(float f) {
  unsigned u = __float_as_uint(f);
  return (u & 0x80000000u) ? ~u : (u | 0x80000000u);
}
__device__ __forceinline__ float ord2f(unsigned u) {
  return __uint_as_float((u & 0x80000000u) ? (u & 0x7FFFFFFFu) : ~u);
}

// -------------------- utility --------------------
__global__ void fill_zero_kernel(float* __restrict__ p, long long n) {
  long long i = (long long)blockIdx.x * blockDim.x + threadIdx.x;
  if (i < n) p[i] = 0.0f;
}

__global__ void degree_kernel(const int* __restrict__ dst, float* __restrict__ deg, int E) {
  int e = blockIdx.x * blockDim.x + threadIdx.x;
  if (e < E) atomicAdd(&deg[dst[e]], 1.0f);
}

// scatter-add x[src] into agg[dst]; one thread per (edge, feature)
__global__ void scatter_add_kernel(const float* __restrict__ x, const int* __restrict__ src,
                                   const int* __restrict__ dst, float* __restrict__ agg,
                                   int E, int F) {
  long long tid = (long long)blockIdx.x * blockDim.x + threadIdx.x;
  if (tid >= (long long)E * F) return;
  int e = (int)(tid / F);
  int f = (int)(tid % F);
  float v = x[(long long)src[e] * F + f];
  atomicAdd(&agg[(long long)dst[e] * F + f], v);
}

// -------------------- WMMA GEMM kernels --------------------
// SAGE layer: out = relu( (agg/deg) @ Wl + xin @ Wr + b )
// block = 128 threads (4 waves), covers a 16-row x 64-col output tile.
// Wl/Wr panels staged through LDS in 64-row K chunks via async copies.
__global__ void __launch_bounds__(128)
sage_gemm_kernel(const float* __restrict__ agg, const float* __restrict__ deg,
                 const float* __restrict__ xin, const float* __restrict__ Wl,
                 const float* __restrict__ Wr, const float* __restrict__ bias,
                 float* __restrict__ out, int K, int N) {
  extern __shared__ float smem[];          // [2][64][LDS_STRIDE]
  float* smL = smem;
  float* smR = smem + 64 * LDS_STRIDE;

  const int lane  = threadIdx.x & 31;
  const int wave  = threadIdx.x >> 5;
  const int half  = lane >> 4;                       // 0: K=0,1  1: K=2,3 inside chunk
  const int r     = blockIdx.x * 16 + (lane & 15);   // A row for this lane
  const int cbase = blockIdx.y * 64;
  const int lc    = wave * 16 + (lane & 15);         // local col 0..63
  const int col   = cbase + lc;
  const float inv = 1.0f / fmaxf(deg[r], 1.0f);

  v8f acc = {};
  for (int kc = 0; kc < K; kc += 64) {
    const int kn = min(64, K - kc);
    // stage Wl/Wr[kc:kc+kn, cbase:cbase+64) into LDS (uniform trip count)
    for (int idx = threadIdx.x; idx < kn * 64; idx += 128) {
      int k = idx >> 6, c = idx & 63;
      async_copy_b32(Wl + (size_t)(kc + k) * N + cbase + c, &smL[k * LDS_STRIDE + c]);
      async_copy_b32(Wr + (size_t)(kc + k) * N + cbase + c, &smR[k * LDS_STRIDE + c]);
    }
    async_wait0();
    __syncthreads();
    for (int k0 = 0; k0 < kn; k0 += 4) {
      const int kk = k0 + 2 * half;
      v2f a, b;
      // chain 1: mean @ Wl
      a.x = agg[(size_t)r * K + kc + kk] * inv;
      a.y = agg[(size_t)r * K + kc + kk + 1] * inv;
      b.x = smL[kk * LDS_STRIDE + lc];
      b.y = smL[(kk + 1) * LDS_STRIDE + lc];
      acc = wmma4(a, b, acc);
      // chain 2: x @ Wr
      a.x = xin[(size_t)r * K + kc + kk];
      a.y = xin[(size_t)r * K + kc + kk + 1];
      b.x = smR[kk * LDS_STRIDE + lc];
      b.y = smR[(kk + 1) * LDS_STRIDE + lc];
      acc = wmma4(a, b, acc);
    }
    __syncthreads();    // protect LDS before next chunk's async writes
  }

  const float bb  = bias[col];
  const int rbase = blockIdx.x * 16 + 8 * half;      // C/D layout: M = v + 8*half
  for (int v = 0; v < 8; ++v) {
    float val = acc[v] + bb;
    out[(size_t)(rbase + v) * N + col] = fmaxf(val, 0.0f);
  }
}

// plain GEMM: out = A @ B (GAT projection), B staged via async->LDS
__global__ void __launch_bounds__(128)
gemm_wmma_kernel(const float* __restrict__ A, const float* __restrict__ B,
                 float* __restrict__ out, int K, int N) {
  extern __shared__ float smem[];          // [64][LDS_STRIDE]

  const int lane  = threadIdx.x & 31;
  const int wave  = threadIdx.x >> 5;
  const int half  = lane >> 4;
  const int r     = blockIdx.x * 16 + (lane & 15);
  const int cbase = blockIdx.y * 64;
  const int lc    = wave * 16 + (lane & 15);
  const int col   = cbase + lc;

  v8f acc = {};
  for (int kc = 0; kc < K; kc += 64) {
    const int kn = min(64, K - kc);
    for (int idx = threadIdx.x; idx < kn * 64; idx += 128) {
      int k = idx >> 6, c = idx & 63;
      async_copy_b32(B + (size_t)(kc + k) * N + cbase + c, &smem[k * LDS_STRIDE + c]);
    }
    async_wait0();
    __syncthreads();
    for (int k0 = 0; k0 < kn; k0 += 4) {
      const int kk = k0 + 2 * half;
      v2f a, b;
      a.x = A[(size_t)r * K + kc + kk];
      a.y = A[(size_t)r * K + kc + kk + 1];
      b.x = smem[kk * LDS_STRIDE + lc];
      b.y = smem[(kk + 1) * LDS_STRIDE + lc];
      acc = wmma4(a, b, acc);
    }
    __syncthreads();
  }
  const int rbase = blockIdx.x * 16 + 8 * half;
  for (int v = 0; v < 8; ++v)
    out[(size_t)(rbase + v) * N + col] = acc[v];
}

// -------------------- GAT attention --------------------
__global__ void gat_scores_kernel(const float* __restrict__ h,
                                  const float* __restrict__ att_src,
                                  const float* __restrict__ att_dst,
                                  float* __restrict__ a_s, float* __restrict__ a_d, int n) {
  int tid = blockIdx.x * blockDim.x + threadIdx.x;
  if (tid >= n * 4) return;
  int node = tid >> 2, hd = tid & 3;
  const float* hp = h + (long long)node * 128 + hd * 32;
  float s = 0.0f, d = 0.0f;
  for (int c = 0; c < 32; ++c) {
    float v = hp[c];
    s += v * att_src[hd * 32 + c];
    d += v * att_dst[hd * 32 + c];
  }
  a_s[tid] = s;
  a_d[tid] = d;
}

__device__ __forceinline__ float edge_logit(const float* a_s, const float* a_d,
                                            int s, int d, int hd) {
  float v = a_s[s * 4 + hd] + a_d[d * 4 + hd];
  return v > 0.0f ? v : 0.2f * v;   // leaky_relu(0.2)
}

__global__ void gat_edge_max_kernel(const int* __restrict__ src, const int* __restrict__ dst,
                                    const float* __restrict__ a_s, const float* __restrict__ a_d,
                                    unsigned* __restrict__ mkey, int E) {
  int tid = blockIdx.x * blockDim.x + threadIdx.x;
  if (tid >= E * 4) return;
  int e = tid >> 2, hd = tid & 3;
  int s = src[e], d = dst[e];
  atomicMax(&mkey[d * 4 + hd], f2ord(edge_logit(a_s, a_d, s, d, hd)));
}

__global__ void gat_edge_expsum_kernel(const int* __restrict__ src, const int* __restrict__ dst,
                                       const float* __restrict__ a_s, const float* __restrict__ a_d,
                                       const unsigned* __restrict__ mkey,
                                       float* __restrict__ ssum, int E) {
  int tid = blockIdx.x * blockDim.x + threadIdx.x;
  if (tid >= E * 4) return;
  int e = tid >> 2, hd = tid & 3;
  int s = src[e], d = dst[e];
  float v = edge_logit(a_s, a_d, s, d, hd);
  float m = ord2f(mkey[d * 4 + hd]);
  atomicAdd(&ssum[d * 4 + hd], __expf(v - m));
}

__global__ void gat_edge_aggregate_kernel(const int* __restrict__ src, const int* __restrict__ dst,
                                          const float* __restrict__ a_s, const float* __restrict__ a_d,
                                          const unsigned* __restrict__ mkey,
                                          const float* __restrict__ ssum,
                                          const float* __restrict__ h,
                                          float* __restrict__ gacc, int E) {
  int tid = blockIdx.x * blockDim.x + threadIdx.x;
  if (tid >= E * 4) return;
  int e = tid >> 2, hd = tid & 3;
  int s = src[e], d = dst[e];
  float v = edge_logit(a_s, a_d, s, d, hd);
  float m = ord2f(mkey[d * 4 + hd]);
  float alpha = __expf(v - m) / (ssum[d * 4 + hd] + 1e-16f);
  const float* hp = h + (long long)s * 128 + hd * 32;
  float* op = gacc + (long long)d * 128 + hd * 32;
  for (int c = 0; c < 32; ++c) atomicAdd(&op[c], alpha * hp[c]);
}

// h2 = elu(gacc + gat_b)
__global__ void elu_bias_kernel(const float* __restrict__ gacc, const float* __restrict__ b,
                                float* __restrict__ h2, long long n, int F) {
  long long tid = (long long)blockIdx.x * blockDim.x + threadIdx.x;
  if (tid >= n) return;
  float v = gacc[tid] + b[tid % F];
  h2[tid] = v > 0.0f ? v : (__expf(v) - 1.0f);
}

// -------------------- classifier --------------------
__global__ void classifier_kernel(const float* __restrict__ h, const float* __restrict__ W,
                                  const float* __restrict__ b, float* __restrict__ out, int n) {
  int node = blockIdx.x * blockDim.x + threadIdx.x;
  if (node >= n) return;
  float acc[5];
  for (int j = 0; j < 5; ++j) acc[j] = b[j];
  const float* hp = h + (long long)node * 256;
  for (int k = 0; k < 256; ++k) {
    float x = hp[k];
    for (int j = 0; j < 5; ++j) acc[j] += x * W[k * 5 + j];
  }
  for (int j = 0; j < 5; ++j) out[(long long)node * 5 + j] = acc[j];
}

// -------------------- launch --------------------
extern "C" void kernel_launch(void* const* d_in, const int* in_sizes, int n_in,
                              void* d_out, int out_size, void* d_ws, size_t ws_size,
                              hipStream_t stream) {
  const float* x        = (const float*)d_in[0];
  const int*   ei       = (const int*)d_in[1];
  const float* s1_Wl    = (const float*)d_in[2];
  const float* s1_Wr    = (const float*)d_in[3];
  const float* s1_b     = (const float*)d_in[4];
  const float* gat_W    = (const float*)d_in[5];
  const float* att_src  = (const float*)d_in[6];
  const float* att_dst  = (const float*)d_in[7];
  const float* gat_b    = (const float*)d_in[8];
  const float* s2_Wl    = (const float*)d_in[9];
  const float* s2_Wr    = (const float*)d_in[10];
  const float* s2_b     = (const float*)d_in[11];
  const float* cls_W    = (const float*)d_in[12];
  const float* cls_b    = (const float*)d_in[13];
  float* out = (float*)d_out;

  const int* src = ei;
  const int* dst = ei + NE;

  // ---- workspace layout (floats) ----
  float* ws = (float*)d_ws;
  float*    deg   = ws;                                       // 50016
  float*    agg1  = deg  + 50016;                             // 50000*20
  unsigned* mkey  = (unsigned*)(agg1 + (long long)NN * 20);   // 50000*4
  float*    ssum  = (float*)mkey + (long long)NN * 4;         // 50000*4
  float*    gacc  = ssum + (long long)NN * 4;                 // 50000*128
  float*    agg2  = gacc + (long long)NN * 128;               // 50000*128
  // --- end of zeroed accumulator region ---
  float*    h1    = agg2 + (long long)NN * 128;               // 50000*256
  float*    hgat  = h1   + (long long)NN * 256;               // 50000*128
  float*    a_s   = hgat + (long long)NN * 128;               // 50000*4
  float*    a_d   = a_s  + (long long)NN * 4;                 // 50000*4
  float*    h2    = a_d  + (long long)NN * 4;                 // 50000*128
  float*    h3    = h2   + (long long)NN * 128;               // 50000*256

  const long long nzero = 50016 + (long long)NN * 20 + (long long)NN * 4 * 2
                        + (long long)NN * 128 * 2;

  const size_t smem_gemm = (size_t)64 * LDS_STRIDE * sizeof(float);      // 16.9 KB
  const size_t smem_sage = 2 * smem_gemm;                                // 33.8 KB

  // 1) zero accumulators (deg, agg1, mkey, ssum, gacc, agg2)
  fill_zero_kernel<<<(unsigned)((nzero + 255) / 256), 256, 0, stream>>>(ws, nzero);

  // 2) degrees
  degree_kernel<<<(NE + 255) / 256, 256, 0, stream>>>(dst, deg, NE);

  // 3) SAGE-1 neighbor sum (F=20)
  {
    long long total = (long long)NE * 20;
    scatter_add_kernel<<<(unsigned)((total + 255) / 256), 256, 0, stream>>>(x, src, dst, agg1, NE, 20);
  }

  // 4) SAGE-1: h1 = relu(mean1@Wl + x@Wr + b)   M=50000 K=20 N=256
  sage_gemm_kernel<<<dim3(NN / 16, 4), 128, smem_sage, stream>>>(agg1, deg, x, s1_Wl, s1_Wr, s1_b, h1, 20, 256);

  // 5) GAT projection: hgat = h1 @ gat_W   K=256 N=128
  gemm_wmma_kernel<<<dim3(NN / 16, 2), 128, smem_gemm, stream>>>(h1, gat_W, hgat, 256, 128);

  // 6) per-node attention scores
  gat_scores_kernel<<<(NN * 4 + 255) / 256, 256, 0, stream>>>(hgat, att_src, att_dst, a_s, a_d, NN);

  // 7-9) per-edge softmax: max, exp-sum, weighted aggregate
  gat_edge_max_kernel<<<(NE * 4 + 255) / 256, 256, 0, stream>>>(src, dst, a_s, a_d, mkey, NE);
  gat_edge_expsum_kernel<<<(NE * 4 + 255) / 256, 256, 0, stream>>>(src, dst, a_s, a_d, mkey, ssum, NE);
  gat_edge_aggregate_kernel<<<(NE * 4 + 255) / 256, 256, 0, stream>>>(src, dst, a_s, a_d, mkey, ssum, hgat, gacc, NE);

  // 10) h2 = elu(gacc + gat_b)
  {
    long long total = (long long)NN * 128;
    elu_bias_kernel<<<(unsigned)((total + 255) / 256), 256, 0, stream>>>(gacc, gat_b, h2, total, 128);
  }

  // 11) SAGE-2 neighbor sum (F=128)
  {
    long long total = (long long)NE * 128;
    scatter_add_kernel<<<(unsigned)((total + 255) / 256), 256, 0, stream>>>(h2, src, dst, agg2, NE, 128);
  }

  // 12) SAGE-2: h3 = relu(mean2@Wl + h2@Wr + b)   K=128 N=256
  sage_gemm_kernel<<<dim3(NN / 16, 4), 128, smem_sage, stream>>>(agg2, deg, h2, s2_Wl, s2_Wr, s2_b, h3, 128, 256);

  // 13) classifier: out = h3 @ cls_W + cls_b
  classifier_kernel<<<(NN + 255) / 256, 256, 0, stream>>>(h3, cls_W, cls_b, out, NN);
}